// DehazeNet_52390011076755
// MI455X (gfx1250) — compile-verified
//
#include <hip/hip_runtime.h>
#include <hip/hip_bf16.h>

typedef __attribute__((ext_vector_type(2))) float v2f;
typedef __attribute__((ext_vector_type(8))) float v8f;

#define TILE  16
#define GP    18           // patches per tile dim incl. halo (16+2)
#define NPAT  (GP*GP)      // 324
#define NPAD  336          // padded to 21*16 rows for WMMA
#define DD    12           // input dim  c*ps*ps
#define NO    36           // output dim 3*c*ps*ps
#define E1    0.36787944117144233f   // exp(-1)
#define E2    0.13533528323661270f   // exp(-2)

// bias[p][py*2+px] from _rel_pos_bias(2)
__device__ __constant__ float c_bias[9][4] = {
  {E2,  E1,  E1,  1.f},   // p0: rot90(br,2)
  {E1,  E1,  1.f, 1.f},   // p1: flip(down,0)
  {E1,  E2,  1.f, E1 },   // p2: rot90(br,1)
  {E1,  1.f, E1,  1.f},   // p3: flip(right,1)
  {1.f, 1.f, 1.f, 1.f},   // p4: ones (center)
  {1.f, E1,  1.f, E1 },   // p5: right
  {E1,  1.f, E2,  E1 },   // p6: rot90(br,3)
  {1.f, 1.f, E1,  E1 },   // p7: down
  {1.f, E1,  E1,  E2 },   // p8: br
};

__global__ __launch_bounds__(256)
void win_attn_fused(const float* __restrict__ x,
                    const float* __restrict__ Wqkv,
                    const float* __restrict__ bqkv,
                    float* __restrict__ out)
{
  __shared__ __align__(16) float xpat[NPAD * DD];   // 16128 B
  __shared__ __align__(16) float qkvs[NPAD * NO];   // 48384 B  (total 64512 < 64KB)

  const int tid    = threadIdx.x;
  const int lane   = tid & 31;
  const int wave   = tid >> 5;
  const int laneN  = lane & 15;
  const int laneHi = lane >> 4;          // 0 or 1

  const int tj = blockIdx.x, ti = blockIdx.y, bb = blockIdx.z;
  const int i0 = ti * TILE, j0 = tj * TILE;

  // ---- Preload B-matrix tiles (B[k][n] = W[n][k]) and bias into registers ----
  float bw[3][3][2];
  float bq[3];
#pragma unroll
  for (int nt = 0; nt < 3; ++nt) {
    const int n   = nt * 16 + laneN;
    const bool ok = (n < NO);
#pragma unroll
    for (int ks = 0; ks < 3; ++ks)
#pragma unroll
      for (int vg = 0; vg < 2; ++vg) {
        const int k = ks * 4 + vg + 2 * laneHi;      // < 12 always
        bw[nt][ks][vg] = ok ? Wqkv[n * DD + k] : 0.0f;
      }
    bq[nt] = ok ? bqkv[n] : 0.0f;
  }

  // ---- Phase A: stage 18x18 patch halo into LDS (zero-pad borders) ----
  for (int idx = tid; idx < NPAD * 6; idx += 256) {
    const int p   = idx / 6;
    const int rem = idx - p * 6;
    const int ch  = rem >> 1, py = rem & 1;
    v2f val = {0.f, 0.f};
    if (p < NPAT) {
      const int pgi = p / GP, pgj = p - pgi * GP;
      const int r  = (i0 + pgi) * 2 - 2 + py;
      const int c0 = (j0 + pgj) * 2 - 2;
      if (r >= 0 && r < 512 && c0 >= 0 && c0 < 512) {
        const float* src = x + (((size_t)bb * 3 + ch) * 512 + r) * 512 + c0;
        val = *(const v2f*)src;                      // 8B aligned (c0 even)
      }
    }
    *(v2f*)(&xpat[p * DD + ch * 4 + py * 2]) = val;
  }

  __syncthreads();

  // ---- Phase B: qkv = xpat[336x12] * W^T[12x36] + b  via v_wmma_f32_16x16x4_f32 ----
  for (int m = wave; m < NPAD / 16; m += 8) {        // wave-uniform trip count
    const int row = m * 16 + laneN;
    const float* ap = &xpat[row * DD + 2 * laneHi];
    v2f a0, a1, a2;
    a0.x = ap[0]; a0.y = ap[1];                      // K 0..3
    a1.x = ap[4]; a1.y = ap[5];                      // K 4..7
    a2.x = ap[8]; a2.y = ap[9];                      // K 8..11
#pragma unroll
    for (int nt = 0; nt < 3; ++nt) {
      v2f b0 = {bw[nt][0][0], bw[nt][0][1]};
      v2f b1 = {bw[nt][1][0], bw[nt][1][1]};
      v2f b2 = {bw[nt][2][0], bw[nt][2][1]};
      v8f acc = {};
      acc = __builtin_amdgcn_wmma_f32_16x16x4_f32(false, a0, false, b0, (short)0, acc, false, false);
      acc = __builtin_amdgcn_wmma_f32_16x16x4_f32(false, a1, false, b1, (short)0, acc, false, false);
      acc = __builtin_amdgcn_wmma_f32_16x16x4_f32(false, a2, false, b2, (short)0, acc, false, false);
      const int n = nt * 16 + laneN;
      if (n < NO) {
#pragma unroll
        for (int r = 0; r < 8; ++r) {
          const int prow = m * 16 + r + 8 * laneHi;  // C/D layout: M = r + 8*lanehalf
          qkvs[prow * NO + n] = acc[r] + bq[nt];
        }
      }
    }
  }

  __syncthreads();

  // ---- Phase C: attention combine + residual ----
  {
    const int li = tid >> 4, lj = tid & 15;
    const int lpc = (li + 1) * GP + (lj + 1);        // center patch

    float qv[12];
    {
      const float4* qb = (const float4*)&qkvs[lpc * NO];
      float4 q0 = qb[0], q1 = qb[1], q2 = qb[2];
      qv[0]=q0.x; qv[1]=q0.y; qv[2]=q0.z; qv[3]=q0.w;
      qv[4]=q1.x; qv[5]=q1.y; qv[6]=q1.z; qv[7]=q1.w;
      qv[8]=q2.x; qv[9]=q2.y; qv[10]=q2.z; qv[11]=q2.w;
    }

    float av[12];
#pragma unroll
    for (int t = 0; t < 12; ++t) av[t] = 0.f;

#pragma unroll
    for (int p = 0; p < 9; ++p) {
      const int di = p / 3, dj = p - di * 3;
      const int lp = (li + di) * GP + (lj + dj);
      const float4* kb = (const float4*)&qkvs[lp * NO + 12];
      const float4* vb = (const float4*)&qkvs[lp * NO + 24];
      float kk[12], vv[12];
      float4 t0 = kb[0], t1 = kb[1], t2 = kb[2];
      kk[0]=t0.x; kk[1]=t0.y; kk[2]=t0.z; kk[3]=t0.w;
      kk[4]=t1.x; kk[5]=t1.y; kk[6]=t1.z; kk[7]=t1.w;
      kk[8]=t2.x; kk[9]=t2.y; kk[10]=t2.z; kk[11]=t2.w;
      float4 u0 = vb[0], u1 = vb[1], u2 = vb[2];
      vv[0]=u0.x; vv[1]=u0.y; vv[2]=u0.z; vv[3]=u0.w;
      vv[4]=u1.x; vv[5]=u1.y; vv[6]=u1.z; vv[7]=u1.w;
      vv[8]=u2.x; vv[9]=u2.y; vv[10]=u2.z; vv[11]=u2.w;
#pragma unroll
      for (int t = 0; t < 12; ++t)
        av[t] = fmaf(qv[t] * kk[t] * c_bias[p][t & 3], vv[t], av[t]);
    }

    const int i = i0 + li, j = j0 + lj;
#pragma unroll
    for (int ch = 0; ch < 3; ++ch)
#pragma unroll
      for (int py = 0; py < 2; ++py) {
        const size_t o = (((size_t)bb * 3 + ch) * 512 + (size_t)(i * 2 + py)) * 512 + (size_t)(j * 2);
        v2f xin = *(const v2f*)(x + o);
        v2f res;
        res.x = xin.x + av[ch * 4 + py * 2 + 0];
        res.y = xin.y + av[ch * 4 + py * 2 + 1];
        *(v2f*)(out + o) = res;
      }
  }
}

extern "C" void kernel_launch(void* const* d_in, const int* in_sizes, int n_in,
                              void* d_out, int out_size, void* d_ws, size_t ws_size,
                              hipStream_t stream) {
  (void)in_sizes; (void)n_in; (void)out_size; (void)d_ws; (void)ws_size;
  const float* x    = (const float*)d_in[0];   // (4,3,512,512) f32
  const float* Wqkv = (const float*)d_in[1];   // (36,12) f32
  const float* bqkv = (const float*)d_in[2];   // (36,) f32
  float* out = (float*)d_out;                  // (4,3,512,512) f32

  dim3 grid(16, 16, 4);                        // 256/16 x 256/16 tiles x batch
  win_attn_fused<<<grid, 256, 0, stream>>>(x, Wqkv, bqkv, out);
}